// SmallestAdditionTransformer_55585466745161
// MI455X (gfx1250) — compile-verified
//
#include <hip/hip_runtime.h>
#include <math.h>

typedef float v2f __attribute__((ext_vector_type(2)));
typedef float v8f __attribute__((ext_vector_type(8)));

#define SEQ_PER_BLOCK 8
#define T_LEN 34
#define ROWS (SEQ_PER_BLOCK * T_LEN)   /* 272 = 17 * 16 */
#define NTILES (ROWS / 16)             /* 17 */

__global__ __launch_bounds__(256)
void tiny_transformer_fused(
    const int*   __restrict__ idx,        // (B, 34)
    const float* __restrict__ tok_emb,    // (14, 3)
    const float* __restrict__ pos_params, // (4,)
    const float* __restrict__ z10_enc,    // (1, 3)
    const float* __restrict__ special_enc,// (3, 3)
    const float* __restrict__ wq,         // (3, 6)
    const float* __restrict__ wk,         // (3, 6)
    const float* __restrict__ wv,         // (3, 6)
    const float* __restrict__ wo,         // (6, 6)
    const float* __restrict__ ln1_g, const float* __restrict__ ln1_b,
    const float* __restrict__ ln2_g, const float* __restrict__ ln2_b,
    const float* __restrict__ lnf_g, const float* __restrict__ lnf_b,
    const float* __restrict__ ffn_w1,     // (6, 2)
    const float* __restrict__ ffn_b1,     // (2,)
    const float* __restrict__ ffn_w2,     // (2, 6)
    const float* __restrict__ ffn_b2,     // (6,)
    const float* __restrict__ head_w,     // (6, 3)
    float* __restrict__ out)              // (B, 34, 14)
{
    __shared__ float sK[ROWS][6];
    __shared__ float sV[ROWS][6];
    __shared__ float sXf[ROWS][8];   // post-lnf activations, cols 6,7 zero-padded
    __shared__ float Wp[8][16];      // head_w @ tok_emb.T, padded 6x14 -> 8x16

    const int tid = threadIdx.x;

    // ---- Wp = head_w @ tok_emb.T (zero-padded) ----
    if (tid < 128) {
        int k = tid >> 4, n = tid & 15;
        float acc = 0.0f;
        if (k < 6 && n < 14) {
            acc = head_w[k*3+0]*tok_emb[n*3+0]
                + head_w[k*3+1]*tok_emb[n*3+1]
                + head_w[k*3+2]*tok_emb[n*3+2];
        }
        Wp[k][n] = acc;
    }

    const float amp   = pos_params[0];
    const float phase = pos_params[1];
    const float slope = pos_params[2];
    const float offs  = pos_params[3];

    float xr[2][6];   // residual stream per handled token
    float qr[2][6];   // queries per handled token

    // ================= Phase A: embed + ln1 + QKV =================
    #pragma unroll
    for (int it = 0; it < 2; ++it) {
        int p = tid + it * 256;
        if (p < ROWS) {
            int s = p / T_LEN;
            int t = p - s * T_LEN;
            long b = (long)blockIdx.x * SEQ_PER_BLOCK + s;
            int tok = idx[b * T_LEN + t];

            float x[6];
            x[0] = tok_emb[tok*3+0];
            x[1] = tok_emb[tok*3+1];
            x[2] = tok_emb[tok*3+2];

            // positional encoding (POS_FLAT logic inlined)
            float p0, p1, p2;
            if (t == 10)      { p0 = special_enc[0]; p1 = special_enc[1]; p2 = special_enc[2]; }
            else if (t == 21) { p0 = special_enc[3]; p1 = special_enc[4]; p2 = special_enc[5]; }
            else if (t == 33) { p0 = special_enc[6]; p1 = special_enc[7]; p2 = special_enc[8]; }
            else if (t == 32) { p0 = z10_enc[0];     p1 = z10_enc[1];     p2 = z10_enc[2];     }
            else {
                int i = (t < 10) ? t : ((t < 21) ? (t - 11) : (t - 22));
                float fi  = (float)i;
                float ang = 0.62831853071795864769f * fi + phase; // 2*pi/10
                p0 = amp * cosf(ang);
                p1 = amp * sinf(ang);
                p2 = slope * fi + offs;
            }
            x[3] = p0; x[4] = p1; x[5] = p2;

            // ln1
            float mu = (x[0]+x[1]+x[2]+x[3]+x[4]+x[5]) * (1.0f/6.0f);
            float var = 0.0f;
            #pragma unroll
            for (int c = 0; c < 6; ++c) { float d = x[c]-mu; var += d*d; }
            var *= (1.0f/6.0f);
            float rs = rsqrtf(var + 1e-5f);
            float h[6];
            #pragma unroll
            for (int c = 0; c < 6; ++c) h[c] = (x[c]-mu)*rs*ln1_g[c] + ln1_b[c];

            // q,k from h_pos=h[3..5]; v from h_tok=h[0..2]
            #pragma unroll
            for (int j = 0; j < 6; ++j) {
                qr[it][j] = h[3]*wq[j] + h[4]*wq[6+j] + h[5]*wq[12+j];
                sK[p][j]  = h[3]*wk[j] + h[4]*wk[6+j] + h[5]*wk[12+j];
                sV[p][j]  = h[0]*wv[j] + h[1]*wv[6+j] + h[2]*wv[12+j];
                xr[it][j] = x[j];
            }
        }
    }
    __syncthreads();

    // ======= Phase B: attention + wo + ln2 + FFN + lnf -> sXf =======
    #pragma unroll
    for (int it = 0; it < 2; ++it) {
        int p = tid + it * 256;
        if (p < ROWS) {
            int s = p / T_LEN;
            int t = p - s * T_LEN;
            int base = s * T_LEN;

            float o[6];
            #pragma unroll
            for (int hh = 0; hh < 2; ++hh) {
                float q0 = qr[it][3*hh+0], q1 = qr[it][3*hh+1], q2 = qr[it][3*hh+2];
                float m = -3.0e38f, den = 0.0f;
                float a0 = 0.0f, a1 = 0.0f, a2 = 0.0f;
                for (int j = 0; j <= t; ++j) {
                    const float* kj = sK[base + j];
                    const float* vj = sV[base + j];
                    float sc = (q0*kj[3*hh+0] + q1*kj[3*hh+1] + q2*kj[3*hh+2]) * 0.57735026918962576f;
                    float nm = fmaxf(m, sc);
                    float corr = __expf(m - nm);
                    float w = __expf(sc - nm);
                    den = den * corr + w;
                    a0 = a0 * corr + w * vj[3*hh+0];
                    a1 = a1 * corr + w * vj[3*hh+1];
                    a2 = a2 * corr + w * vj[3*hh+2];
                    m = nm;
                }
                float inv = 1.0f / den;
                o[3*hh+0] = a0 * inv;
                o[3*hh+1] = a1 * inv;
                o[3*hh+2] = a2 * inv;
            }

            // x = x + o @ wo
            float x2[6];
            #pragma unroll
            for (int j = 0; j < 6; ++j) {
                float acc = xr[it][j];
                #pragma unroll
                for (int c = 0; c < 6; ++c) acc += o[c] * wo[c*6 + j];
                x2[j] = acc;
            }

            // ln2
            float mu = (x2[0]+x2[1]+x2[2]+x2[3]+x2[4]+x2[5]) * (1.0f/6.0f);
            float var = 0.0f;
            #pragma unroll
            for (int c = 0; c < 6; ++c) { float d = x2[c]-mu; var += d*d; }
            var *= (1.0f/6.0f);
            float rs = rsqrtf(var + 1e-5f);
            float h2[6];
            #pragma unroll
            for (int c = 0; c < 6; ++c) h2[c] = (x2[c]-mu)*rs*ln2_g[c] + ln2_b[c];

            // FFN: gelu(h2 @ w1 + b1) @ w2 + b2, exact erf gelu
            float f0 = ffn_b1[0], f1 = ffn_b1[1];
            #pragma unroll
            for (int j = 0; j < 6; ++j) {
                f0 += h2[j] * ffn_w1[j*2+0];
                f1 += h2[j] * ffn_w1[j*2+1];
            }
            f0 = 0.5f * f0 * (1.0f + erff(f0 * 0.70710678118654752f));
            f1 = 0.5f * f1 * (1.0f + erff(f1 * 0.70710678118654752f));
            #pragma unroll
            for (int j = 0; j < 6; ++j)
                x2[j] += f0 * ffn_w2[j] + f1 * ffn_w2[6+j] + ffn_b2[j];

            // lnf -> sXf (pad cols 6,7 with zeros for WMMA K-padding)
            mu = (x2[0]+x2[1]+x2[2]+x2[3]+x2[4]+x2[5]) * (1.0f/6.0f);
            var = 0.0f;
            #pragma unroll
            for (int c = 0; c < 6; ++c) { float d = x2[c]-mu; var += d*d; }
            var *= (1.0f/6.0f);
            rs = rsqrtf(var + 1e-5f);
            #pragma unroll
            for (int c = 0; c < 6; ++c) sXf[p][c] = (x2[c]-mu)*rs*lnf_g[c] + lnf_b[c];
            sXf[p][6] = 0.0f;
            sXf[p][7] = 0.0f;
        }
    }
    __syncthreads();

    // ====== Phase C: logits = sXf(272x8) @ Wp(8x16) via WMMA f32 16x16x4 ======
    const int wave = tid >> 5;
    const int lane = tid & 31;
    const int nrow  = lane & 15;            // A row / B col / D col index
    const int khalf = (lane >> 4) << 1;     // 0 for lanes 0-15, 2 for lanes 16-31
    const int mofs  = (lane >> 4) * 8;      // D: M offset per lane half
    float* outBlk = out + (size_t)blockIdx.x * (size_t)ROWS * 14;

    for (int tile = wave; tile < NTILES; tile += 8) {   // wave-uniform: EXEC stays full
        int row = tile * 16 + nrow;
        v2f a1, a2, b1, b2;
        // A 16x4 f32 layout: VGPR0 -> K=khalf, VGPR1 -> K=khalf+1
        a1.x = sXf[row][khalf];     a1.y = sXf[row][khalf + 1];
        a2.x = sXf[row][4 + khalf]; a2.y = sXf[row][5 + khalf];
        // B 4x16 f32 layout (row striped across lanes): VGPR0 -> K=khalf, VGPR1 -> K=khalf+1
        b1.x = Wp[khalf][nrow];     b1.y = Wp[khalf + 1][nrow];
        b2.x = Wp[4 + khalf][nrow]; b2.y = Wp[5 + khalf][nrow];

        v8f c = {0.f,0.f,0.f,0.f,0.f,0.f,0.f,0.f};
        v8f d = __builtin_amdgcn_wmma_f32_16x16x4_f32(false, a1, false, b1, (short)0, c, false, false);
        d     = __builtin_amdgcn_wmma_f32_16x16x4_f32(false, a2, false, b2, (short)0, d, false, false);

        // D 16x16 f32: VGPR r -> M = r (lanes 0-15) / r+8 (lanes 16-31), N = lane&15
        if (nrow < 14) {
            #pragma unroll
            for (int r = 0; r < 8; ++r) {
                int m = tile * 16 + r + mofs;
                outBlk[(size_t)m * 14 + nrow] = d[r];
            }
        }
    }
}

extern "C" void kernel_launch(void* const* d_in, const int* in_sizes, int n_in,
                              void* d_out, int out_size, void* d_ws, size_t ws_size,
                              hipStream_t stream) {
    (void)in_sizes; (void)n_in; (void)out_size; (void)d_ws; (void)ws_size;
    const int*   idx         = (const int*)  d_in[0];
    const float* tok_emb     = (const float*)d_in[1];
    const float* pos_params  = (const float*)d_in[2];
    const float* z10_enc     = (const float*)d_in[3];
    const float* special_enc = (const float*)d_in[4];
    const float* wq          = (const float*)d_in[5];
    const float* wk          = (const float*)d_in[6];
    const float* wv          = (const float*)d_in[7];
    const float* wo          = (const float*)d_in[8];
    const float* ln1_g       = (const float*)d_in[9];
    const float* ln1_b       = (const float*)d_in[10];
    const float* ln2_g       = (const float*)d_in[11];
    const float* ln2_b       = (const float*)d_in[12];
    const float* lnf_g       = (const float*)d_in[13];
    const float* lnf_b       = (const float*)d_in[14];
    const float* ffn_w1      = (const float*)d_in[15];
    const float* ffn_b1      = (const float*)d_in[16];
    const float* ffn_w2      = (const float*)d_in[17];
    const float* ffn_b2      = (const float*)d_in[18];
    const float* head_w      = (const float*)d_in[19];
    float* out = (float*)d_out;

    const int B = 32768;
    dim3 grid(B / SEQ_PER_BLOCK);   // 4096 blocks, 8 sequences each
    dim3 block(256);                // 8 wave32s
    tiny_transformer_fused<<<grid, block, 0, stream>>>(
        idx, tok_emb, pos_params, z10_enc, special_enc,
        wq, wk, wv, wo,
        ln1_g, ln1_b, ln2_g, ln2_b, lnf_g, lnf_b,
        ffn_w1, ffn_b1, ffn_w2, ffn_b2, head_w, out);
}